// ExpMovingAverage_73160472920665
// MI455X (gfx1250) — compile-verified
//
#include <hip/hip_runtime.h>

// ---------------------------------------------------------------------------
// EMA y[t] = a*x[t] + (1-a)*y[t-1], a=0.3, y[0]=x[0], B=512 rows, T=65536.
//
// 0.7^48 ~ 3.7e-8 < fp32 ulp  =>  the IIR is numerically a 48-tap FIR:
//   y[t] = sum_{k=0}^{47} (0.3*0.7^k) * x[t-k]   (+ 0.7^{t+1}*x[0] for t<48)
// i.e. a banded-Toeplitz matmul -> V_WMMA_F32_16X16X4_F32 (fp32 exact).
// Each wave: 16 chained k=4 WMMAs -> one 16x16 f32 D tile = 256 consecutive
// outputs of one row. Memory-bound: ~268 MB total -> ~11.5 us HBM floor.
//
// v3: reversed weight table (A pair loads in ascending order -> no swap movs);
//     float4 (b128) global->LDS tile load, scalar path only for the 64 halo.
// ---------------------------------------------------------------------------

typedef __attribute__((ext_vector_type(2))) float v2f;
typedef __attribute__((ext_vector_type(8))) float v8f;

#define TAPS   48          // truncation window (0.7^48 below fp32 ulp)
#define AW     64          // A-matrix width = TAPS + 16 (padded band)
#define WTN    80          // reversed, zero-padded weight table size
#define SEG    2048        // outputs per workgroup (8 waves * 256)
#define HALO   64          // LDS halo (>= TAPS, 16-aligned)
#define NTHR   256

__global__ __launch_bounds__(NTHR)
void ema_wmma_f32_kernel(const float* __restrict__ x,
                         float* __restrict__ y,
                         int T) {
    __shared__ float xl[SEG + HALO]; // input tile + left halo
    __shared__ float rev[WTN];       // rev[m] = 0.3*0.7^(63-m) if 63-m in [0,48)
    __shared__ float ptab[AW];       // ptab[t] = 0.7^{t+1} (start correction)

    const int  tid     = threadIdx.x;
    const int  row     = blockIdx.y;
    const long segBase = (long)blockIdx.x * SEG;
    const float* __restrict__ xr = x + (long)row * T;

    // Build weight tables once per block (fp64 accumulation -> exact fp32 taps).
    if (tid == 0) {
        for (int i = 0; i < WTN; ++i) rev[i] = 0.0f;
        double p = 1.0;
        for (int k = 0; k < AW; ++k) {
            if (k < TAPS) rev[63 - k] = (float)(0.3 * p);
            ptab[k] = (float)(0.7 * p);   // 0.7^{k+1}
            p *= 0.7;
        }
    }
    // Main tile: 16B-aligned, always in-bounds -> b128 copies.
    {
        const float4* __restrict__ xr4 = (const float4*)(xr + segBase);
        float4* xl4 = (float4*)(xl + HALO);
        for (int i = tid; i < SEG / 4; i += NTHR) xl4[i] = xr4[i];
    }
    // Left halo (64 floats), zero-padded before t=0.
    if (tid < HALO) {
        long g = segBase - HALO + tid;
        xl[tid] = (g >= 0) ? xr[g] : 0.0f;
    }
    __syncthreads();

    const int wave    = tid >> 5;
    const int lane    = tid & 31;
    const int r       = lane & 15;    // A: row M ; B/D: column N
    const int hi      = lane >> 4;    // K-pair / M-half selector
    const int waveOff = wave * 256;   // this wave's 256 outputs within segment

    // A[r][j] = w[r+48-j] in band = rev[(15-r)+j]; ascending adjacent pairs.
    const int aBase = 15 - r;
    // B[j][n] = xl[HALO + waveOff + 16*n - TAPS + j], n = r, j = 0..AW-1
    const int bBase = HALO + waveOff + 16 * r - TAPS;

    // 16 chained 16x16x4 f32 WMMAs (uniform control flow, EXEC all-1s).
    v8f acc = {};
#pragma unroll
    for (int kk = 0; kk < 16; ++kk) {
        const int j0 = 4 * kk + 2 * hi;
        v2f a, b;
        a.x = rev[aBase + j0];        // A[r][j0]
        a.y = rev[aBase + j0 + 1];    // A[r][j0+1]
        b.x = xl[bBase + j0];
        b.y = xl[bBase + j0 + 1];
        // (neg_a, A, neg_b, B, c_mod, C, reuse_a, reuse_b)
        acc = __builtin_amdgcn_wmma_f32_16x16x4_f32(
                  false, a, false, b, (short)0, acc, false, false);
    }

    // D layout: lane L, VGPR v  ->  t = base + 16*(L&15) + 8*(L>>4) + v
    const long outT = segBase + waveOff + 16 * r + 8 * hi;

    float cv[8];
#pragma unroll
    for (int v = 0; v < 8; ++v) cv[v] = acc[v];

    // Sequence-start correction: y[t] += 0.7^{t+1} * x[0] for global t < TAPS.
    if (blockIdx.x == 0 && wave == 0) {
        const float x0 = xl[HALO];             // x[row][0]
#pragma unroll
        for (int v = 0; v < 8; ++v) {
            long t = outT + v;
            if (t < TAPS) cv[v] += ptab[t] * x0;
        }
    }

    float* __restrict__ yo = y + (long)row * T + outT;  // 32B aligned
    *(float4*)(yo)     = make_float4(cv[0], cv[1], cv[2], cv[3]);
    *(float4*)(yo + 4) = make_float4(cv[4], cv[5], cv[6], cv[7]);
}

extern "C" void kernel_launch(void* const* d_in, const int* in_sizes, int n_in,
                              void* d_out, int out_size, void* d_ws, size_t ws_size,
                              hipStream_t stream) {
    (void)n_in; (void)d_ws; (void)ws_size; (void)out_size;
    const float* x = (const float*)d_in[0];
    float*       y = (float*)d_out;

    const int T  = 65536;
    const int Bn = in_sizes[0] / T;            // 512

    dim3 grid(T / SEG, Bn);                    // 32 x 512 workgroups
    ema_wmma_f32_kernel<<<grid, NTHR, 0, stream>>>(x, y, T);
}